// SRGNNConv_30751965840098
// MI455X (gfx1250) — compile-verified
//
#include <hip/hip_runtime.h>

typedef __attribute__((ext_vector_type(2))) float v2f;
typedef __attribute__((ext_vector_type(8))) float v8f;

#define DIM 100
#define NTILES_D 7   // ceil(100/16)

// ---------------------------------------------------------------------------
// Kernel 1: zero the output accumulator (harness poisons d_out with 0xAA).
// out_size = 50000*100, divisible by 4 -> float4 stores.
// ---------------------------------------------------------------------------
__global__ void srgnn_zero_kernel(float4* __restrict__ out, int n4) {
  int i = blockIdx.x * blockDim.x + threadIdx.x;
  if (i < n4) {
    float4 z; z.x = 0.f; z.y = 0.f; z.z = 0.f; z.w = 0.f;
    out[i] = z;
  }
}

// ---------------------------------------------------------------------------
// Kernel 2: hidden = ego @ W^T + b  via V_WMMA_F32_16X16X4_F32.
// One wave32 per (16-row, 16-col) output tile. K=100 -> 25 steps of K=4.
//
// Fragment layouts (CDNA5 ISA 7.12.2, fp32 16x16x4):
//   A (16x4):  lanes 0-15 hold M=lane, VGPR{0,1}=K{0,1}; lanes 16-31: K{2,3}
//   B (4x16):  lanes 0-15 hold N=lane, VGPR{0,1}=K{0,1}; lanes 16-31: K{2,3}
//   C/D (16x16): lane<16: VGPRv = D[v][lane]; lane>=16: VGPRv = D[v+8][lane-16]
// ---------------------------------------------------------------------------
__global__ void srgnn_gemm_wmma_kernel(const float* __restrict__ ego,
                                       const float* __restrict__ W,
                                       const float* __restrict__ bias,
                                       float* __restrict__ hidden,
                                       int n_nodes, int total_tasks) {
  const int wave = (blockIdx.x * blockDim.x + threadIdx.x) >> 5;
  if (wave >= total_tasks) return;        // wave-uniform: EXEC stays all-1s

  const int lane = threadIdx.x & 31;
  const int half = lane >> 4;             // 0: lanes 0-15, 1: lanes 16-31
  const int lr   = lane & 15;

  const int mtile = wave / NTILES_D;
  const int ntile = wave % NTILES_D;
  const int m0 = mtile * 16;
  const int d0 = ntile * 16;

  const int  arow_i  = m0 + lr;
  const int  arow_c  = arow_i < n_nodes ? arow_i : (n_nodes - 1);
  const int  d       = d0 + lr;
  const bool dvalid  = d < DIM;

  const float* __restrict__ arow = ego + (size_t)arow_c * DIM;
  const float* __restrict__ brow = W + (size_t)(dvalid ? d : 0) * DIM; // B[k][d]=W[d][k]

  v8f acc = {};
#pragma unroll
  for (int kk = 0; kk < 25; ++kk) {
    const int k = kk * 4 + half * 2;
    v2f af, bf;
    af.x = arow[k];
    af.y = arow[k + 1];
    const float bx = brow[k];
    const float by = brow[k + 1];
    bf.x = dvalid ? bx : 0.0f;            // VALU select, EXEC untouched
    bf.y = dvalid ? by : 0.0f;
    acc = __builtin_amdgcn_wmma_f32_16x16x4_f32(
        /*neg_a=*/false, af, /*neg_b=*/false, bf,
        /*c_mod=*/(short)0, acc, /*reuse_a=*/false, /*reuse_b=*/false);
  }

  if (dvalid) {
    const float bv = bias[d];
#pragma unroll
    for (int v = 0; v < 8; ++v) {
      const int m = m0 + v + 8 * half;
      if (m < n_nodes)
        hidden[(size_t)m * DIM + d] = acc[v] + bv;
    }
  }
}

// ---------------------------------------------------------------------------
// Kernel 3: per-edge gather * weight, atomic scatter-add into out[dst].
// One wave32 per edge (grid-stride). hidden & out are L2-resident (40 MB
// total vs 192 MB L2), so this runs at L2 atomic throughput.
// DIM = 100 = 3*32 + 4 lanes.
// ---------------------------------------------------------------------------
__global__ void srgnn_scatter_kernel(const float* __restrict__ hidden,
                                     const int* __restrict__ esrc,
                                     const int* __restrict__ edst,
                                     const float* __restrict__ ewgt,
                                     float* __restrict__ out,
                                     int n_edges) {
  const int lane   = threadIdx.x & 31;
  const int wave   = (blockIdx.x * blockDim.x + threadIdx.x) >> 5;
  const int nwaves = (gridDim.x * blockDim.x) >> 5;

  for (int e = wave; e < n_edges; e += nwaves) {
    const int   s = esrc[e];
    const int   d = edst[e];
    const float w = ewgt[e];

    const float* __restrict__ hrow = hidden + (size_t)s * DIM;
    float* __restrict__       orow = out + (size_t)d * DIM;

    // Prefetch next edge's source row (global_prefetch_b8) — hides the
    // dependent-load latency of the gather chain.
    const int en = e + nwaves;
    if (en < n_edges) {
      const float* nrow = hidden + (size_t)esrc[en] * DIM;
      __builtin_prefetch(nrow + lane, 0, 1);
    }

#pragma unroll
    for (int it = 0; it < 3; ++it) {
      const int c = it * 32 + lane;
      atomicAdd(orow + c, hrow[c] * w);   // global_atomic_add_f32, no return
    }
    if (lane < 4) {
      const int c = 96 + lane;
      atomicAdd(orow + c, hrow[c] * w);
    }
  }
}

// ---------------------------------------------------------------------------
extern "C" void kernel_launch(void* const* d_in, const int* in_sizes, int n_in,
                              void* d_out, int out_size, void* d_ws, size_t ws_size,
                              hipStream_t stream) {
  const float* ego  = (const float*)d_in[0];
  const int*   esrc = (const int*)  d_in[1];
  const int*   edst = (const int*)  d_in[2];
  const float* ewgt = (const float*)d_in[3];
  const float* W    = (const float*)d_in[4];
  const float* bias = (const float*)d_in[5];
  float*       out  = (float*)d_out;
  float*       hidden = (float*)d_ws;     // 50000*100*4 = 20 MB scratch

  const int n_nodes = in_sizes[0] / DIM;  // 50000
  const int n_edges = in_sizes[1];        // 800000

  // 1) zero the output accumulator
  {
    const int n4 = out_size / 4;
    const int blk = 256;
    srgnn_zero_kernel<<<(n4 + blk - 1) / blk, blk, 0, stream>>>(
        (float4*)out, n4);
  }

  // 2) fp32 WMMA projection into workspace
  {
    const int mtiles = (n_nodes + 15) / 16;        // 3125
    const int tasks  = mtiles * NTILES_D;          // 21875 wave-tiles
    const int waves_per_block = 8;                 // 256 threads
    const int blocks = (tasks + waves_per_block - 1) / waves_per_block;
    srgnn_gemm_wmma_kernel<<<blocks, 256, 0, stream>>>(
        ego, W, bias, hidden, n_nodes, tasks);
  }

  // 3) gather / weight / atomic scatter-add
  {
    const int blocks = 16384;                      // 131072 waves, grid-stride
    srgnn_scatter_kernel<<<blocks, 256, 0, stream>>>(
        hidden, esrc, edst, ewgt, out, n_edges);
  }
}